// MyFunctionBlock_43276090474802
// MI455X (gfx1250) — compile-verified
//
#include <hip/hip_runtime.h>
#include <hip/hip_bf16.h>
#include <math.h>

// ---------------------------------------------------------------------------
// FunctionBlock (sin): out[b] = { xc.w_sum + b_sum , exp(log|xc|.w_prod) }
// with xc = [x, sin(x)], D=64 -> 128 features.  HBM-bound (~277MB @23.3TB/s
// => ~12us floor).  One wave32 per 16-row tile; both per-row reductions run
// as V_WMMA_F32_16X16X4_F32 K-chains so the matrix pipe does the dot products
// while the VALU does the sin/log transcendentals.
//
// VALU diet vs round 1: raw v_log_f32 (|x| src modifier, no denorm guard) and
// the identity exp(sum w*ln|x|) == exp2(sum w*log2|x|) remove every ln2/log2e
// fixup -> loop body drops from ~800 to ~220 instrs per 4KB tile, putting the
// kernel back under the HBM roofline instead of the issue-rate roofline.
// ---------------------------------------------------------------------------

typedef float v2f __attribute__((ext_vector_type(2)));
typedef float v8f __attribute__((ext_vector_type(8)));

#define WMMA_ACC(acc, a, b)                                                   \
  acc = __builtin_amdgcn_wmma_f32_16x16x4_f32(false, (a), false, (b),         \
                                              (short)0, (acc), false, false)

// Single v_log_f32 with |.| source modifier; inputs are N(0,1) activations,
// denormals are probability-zero, so skip libm's denorm prescale path.
__device__ __forceinline__ float fast_log2_abs(float x) {
#if __has_builtin(__builtin_amdgcn_logf)
  return __builtin_amdgcn_logf(__builtin_fabsf(x));
#else
  return __log2f(fabsf(x));
#endif
}

__device__ __forceinline__ float fast_exp2(float x) {
#if __has_builtin(__builtin_amdgcn_exp2f)
  return __builtin_amdgcn_exp2f(x);
#else
  return exp2f(x);
#endif
}

__device__ __forceinline__ float v8_get(v8f v, int i) {
  float r = v[0];
  r = (i == 1) ? v[1] : r;
  r = (i == 2) ? v[2] : r;
  r = (i == 3) ? v[3] : r;
  r = (i == 4) ? v[4] : r;
  r = (i == 5) ? v[5] : r;
  r = (i == 6) ? v[6] : r;
  r = (i == 7) ? v[7] : r;
  return r;
}

__global__ void __launch_bounds__(256)
fnblock_sin_sum_prod_kernel(const float* __restrict__ x,
                            const float* __restrict__ w_sum,
                            const float* __restrict__ b_sum,
                            const float* __restrict__ w_prod,
                            float* __restrict__ out,
                            int ntiles) {
  const int lane = threadIdx.x & 31;
  const int h    = lane >> 4;   // half-wave: supplies K={0,1} (h=0) or K={2,3} (h=1)
  const int m    = lane & 15;   // A-matrix row M within the 16-row tile

  const int wavesPerBlock = blockDim.x >> 5;
  const int waveId  = blockIdx.x * wavesPerBlock + (threadIdx.x >> 5);
  const int nWaves  = gridDim.x * wavesPerBlock;

  // --- Hoist B-fragments (weights) into registers, once per wave ---------
  // K-order permutation: WMMA step kb, K-slot (2h+t) <-> feature 32h+2kb+t.
  // All 16 B-columns carry the same pair, so every lane's C/D holds the full
  // per-row result.  w_prod is used raw: log2/exp2 absorb the ln2 factors.
  v2f ws0[16], ws1[16], wp0[16], wp1[16];
  {
    const float* p0 = w_sum  + 32 * h;
    const float* p1 = w_sum  + 64 + 32 * h;
    const float* q0 = w_prod + 32 * h;
    const float* q1 = w_prod + 64 + 32 * h;
#pragma unroll
    for (int kb = 0; kb < 16; ++kb) {
      ws0[kb] = *(const v2f*)(p0 + 2 * kb);
      ws1[kb] = *(const v2f*)(p1 + 2 * kb);
      wp0[kb] = *(const v2f*)(q0 + 2 * kb);
      wp1[kb] = *(const v2f*)(q1 + 2 * kb);
    }
  }
  const float bias = b_sum[0];

  for (int tile = waveId; tile < ntiles; tile += nWaves) {
    const int row0 = tile << 4;
    // Each lane loads its contiguous half-row (32 floats) with 8x b128.
    const float* rp = x + (size_t)(row0 + m) * 64 + 32 * h;
    float4 xv[8];
#pragma unroll
    for (int j = 0; j < 8; ++j) xv[j] = *(const float4*)(rp + 4 * j);

    // Two independent chains per result (x-part / sin-part) halve the
    // dependent-WMMA critical path; merged with one add at the end.
    v8f accS0 = {0.f, 0.f, 0.f, 0.f, 0.f, 0.f, 0.f, 0.f};
    v8f accS1 = {0.f, 0.f, 0.f, 0.f, 0.f, 0.f, 0.f, 0.f};
    v8f accP0 = {0.f, 0.f, 0.f, 0.f, 0.f, 0.f, 0.f, 0.f};
    v8f accP1 = {0.f, 0.f, 0.f, 0.f, 0.f, 0.f, 0.f, 0.f};

    // ---- sum chain, x part (features 0..63) ----
#pragma unroll
    for (int j = 0; j < 8; ++j) {
      v2f a0 = {xv[j].x, xv[j].y};
      v2f a1 = {xv[j].z, xv[j].w};
      WMMA_ACC(accS0, a0, ws0[2 * j]);
      WMMA_ACC(accS0, a1, ws0[2 * j + 1]);
    }

    // ---- sin(x): reuses loaded registers, no extra memory traffic ----
    float4 sv[8];
#pragma unroll
    for (int j = 0; j < 8; ++j) {
      sv[j].x = __sinf(xv[j].x);
      sv[j].y = __sinf(xv[j].y);
      sv[j].z = __sinf(xv[j].z);
      sv[j].w = __sinf(xv[j].w);
    }

    // ---- sum chain, sin part (features 64..127) ----
#pragma unroll
    for (int j = 0; j < 8; ++j) {
      v2f a0 = {sv[j].x, sv[j].y};
      v2f a1 = {sv[j].z, sv[j].w};
      WMMA_ACC(accS1, a0, ws1[2 * j]);
      WMMA_ACC(accS1, a1, ws1[2 * j + 1]);
    }

    // ---- prod chain: log2|x| part (raw v_log_f32, |.| modifier) ----
#pragma unroll
    for (int j = 0; j < 8; ++j) {
      xv[j].x = fast_log2_abs(xv[j].x);
      xv[j].y = fast_log2_abs(xv[j].y);
      xv[j].z = fast_log2_abs(xv[j].z);
      xv[j].w = fast_log2_abs(xv[j].w);
    }
#pragma unroll
    for (int j = 0; j < 8; ++j) {
      v2f a0 = {xv[j].x, xv[j].y};
      v2f a1 = {xv[j].z, xv[j].w};
      WMMA_ACC(accP0, a0, wp0[2 * j]);
      WMMA_ACC(accP0, a1, wp0[2 * j + 1]);
    }

    // ---- prod chain: log2|sin(x)| part ----
#pragma unroll
    for (int j = 0; j < 8; ++j) {
      sv[j].x = fast_log2_abs(sv[j].x);
      sv[j].y = fast_log2_abs(sv[j].y);
      sv[j].z = fast_log2_abs(sv[j].z);
      sv[j].w = fast_log2_abs(sv[j].w);
    }
#pragma unroll
    for (int j = 0; j < 8; ++j) {
      v2f a0 = {sv[j].x, sv[j].y};
      v2f a1 = {sv[j].z, sv[j].w};
      WMMA_ACC(accP1, a0, wp1[2 * j]);
      WMMA_ACC(accP1, a1, wp1[2 * j + 1]);
    }

    // ---- emit: lanes 0-15 carry rows 0..7, lanes 16-31 rows 8..15;
    // exp(sum w*ln|x|) == exp2(sum w*log2|x|) -> single v_exp_f32.
    if (m < 8) {
      float s = v8_get(accS0, m) + v8_get(accS1, m) + bias;
      float p = fast_exp2(v8_get(accP0, m) + v8_get(accP1, m));
      const int row = row0 + 8 * h + m;
      float2 o;
      o.x = s;
      o.y = p;
      *(float2*)(out + (size_t)row * 2) = o;
    }
  }
}

extern "C" void kernel_launch(void* const* d_in, const int* in_sizes, int n_in,
                              void* d_out, int out_size, void* d_ws, size_t ws_size,
                              hipStream_t stream) {
  const float* x      = (const float*)d_in[0];
  const float* w_sum  = (const float*)d_in[1];
  const float* b_sum  = (const float*)d_in[2];
  const float* w_prod = (const float*)d_in[3];
  float* out = (float*)d_out;

  const int B = in_sizes[0] / 64;   // rows
  const int ntiles = B / 16;        // 16 rows per wave-tile (B = 1M -> 65536)

  const int threads = 256;                       // 8 wave32 per block
  const int wavesPerBlock = threads / 32;
  const int tilesPerWave = 4;                    // amortize weight-register setup
  int blocks = (ntiles + wavesPerBlock * tilesPerWave - 1) /
               (wavesPerBlock * tilesPerWave);
  if (blocks < 1) blocks = 1;

  fnblock_sin_sum_prod_kernel<<<blocks, threads, 0, stream>>>(
      x, w_sum, b_sum, w_prod, out, ntiles);
}